// SpatialGraphAttentionLayer_83605833384122
// MI455X (gfx1250) — compile-verified
//
#include <hip/hip_runtime.h>
#include <math.h>

#define BB 2
#define NN 10000
#define FIN 128
#define EE 160000
#define WEDGE 16
#define HEADS 4
#define HEAD_DIM 32
#define INNER 128
#define OUTD 128
#define NEG_SLOPE 0.2f
#define LN_EPS 1e-5f

typedef __attribute__((ext_vector_type(2))) float v2f;
typedef __attribute__((ext_vector_type(8))) float v8f;

// nan_to_num(nan=0, posinf=0, neginf=0)
__device__ __forceinline__ float san(float v) {
  return __builtin_isfinite(v) ? v : 0.0f;
}

// order-preserving float<->uint map for atomic segment-max
__device__ __forceinline__ unsigned fenc(float f) {
  unsigned b = __float_as_uint(f);
  return b ^ ((unsigned)((int)b >> 31) | 0x80000000u);
}
__device__ __forceinline__ float fdec(unsigned u) {
  unsigned b = (u & 0x80000000u) ? (u ^ 0x80000000u) : ~u;
  return __uint_as_float(b);
}
#define ENC_NEGINF 0x007FFFFFu  // fenc(-inf)

// ---- CDNA5 async global->LDS copy (ASYNCcnt-tracked, bypasses VGPRs) ----
// global_load_async_to_lds_b128 vdst(LDS byte addr), v[addr64], off
__device__ __forceinline__ void async_copy_b128(unsigned lds_off, const void* gaddr) {
  asm volatile("global_load_async_to_lds_b128 %0, %1, off"
               :: "v"(lds_off), "v"((unsigned long long)(uintptr_t)gaddr)
               : "memory");
}
__device__ __forceinline__ void async_wait0() {
  asm volatile("s_wait_asynccnt 0x0" ::: "memory");
}

// Wave-level 16x16 fp32 GEMM tile: C(16x16) = A(16x128, LDS row-major) x B,
// where B[k][n] = W[n*128 + k] (i.e. multiply by W^T). K=128 in 32 steps of
// V_WMMA_F32_16X16X4_F32. wave32: A frag = 2 VGPRs, B frag = 2 VGPRs, D = 8.
__device__ __forceinline__ v8f wmma_tile(const float* As, const float* __restrict__ W,
                                         int ncol0, int lane) {
  const int m  = lane & 15;
  const int hf = lane >> 4;      // which 16-lane half
  const int n  = ncol0 + m;      // B/D column index for this lane
  v8f acc = {};
  for (int k0 = 0; k0 < 128; k0 += 4) {
    const int ka = k0 + 2 * hf;  // A VGPR0:{K=0,K=2}, VGPR1:{K=1,K=3} layout
    v2f a, b;
    a.x = As[m * 128 + ka];
    a.y = As[m * 128 + ka + 1];
    b.x = W[(size_t)n * 128 + ka];
    b.y = W[(size_t)n * 128 + ka + 1];
    acc = __builtin_amdgcn_wmma_f32_16x16x4_f32(false, a, false, b, (short)0,
                                                acc, false, false);
  }
  return acc;
}

// ---------------------------------------------------------------- init
__global__ void k_init(float* __restrict__ agg, float* __restrict__ den,
                       unsigned* __restrict__ mseg) {
  const size_t i = (size_t)blockIdx.x * blockDim.x + threadIdx.x;
  if (i < (size_t)BB * NN * INNER) agg[i] = 0.0f;
  if (i < (size_t)NN * BB * HEADS) { den[i] = 0.0f; mseg[i] = ENC_NEGINF; }
}

// ---------------------------------------------- h = x @ Wn^T  (+ attn pre-dots)
__global__ void k_node_transform(const float* __restrict__ x,
                                 const float* __restrict__ Wn,
                                 const float* __restrict__ attn,
                                 float* __restrict__ h,
                                 float* __restrict__ ps,
                                 float* __restrict__ pd) {
  __shared__ float xs[16 * FIN];
  __shared__ float ht[16 * INNER];
  const int row0 = blockIdx.x * 16;   // flat row index into (B*N)
  const int tid  = threadIdx.x;
  for (int i = 0; i < 8; ++i) {
    int li = tid + 256 * i;           // 16*128 = 2048 elements
    xs[li] = san(x[(size_t)row0 * FIN + li]);   // nan_to_num applied in flight
  }
  __syncthreads();
  const int wave = tid >> 5, lane = tid & 31;
  v8f acc = wmma_tile(xs, Wn, wave * 16, lane);
  const int hf  = lane >> 4;
  const int col = wave * 16 + (lane & 15);
  for (int v = 0; v < 8; ++v) {
    int rm = v + 8 * hf;              // C/D layout: lanes16-31 hold M=v+8
    float d = acc[v];
    h[((size_t)row0 + rm) * INNER + col] = d;
    ht[rm * INNER + col] = d;
  }
  __syncthreads();
  // p_s[bn][hd] = h[bn, hd*32:+32] . a_s[hd] ; same for p_d with a_d
  if (tid < 16 * HEADS) {
    int r = tid >> 2, hd = tid & 3;
    float s1 = 0.0f, s2 = 0.0f;
    for (int d = 0; d < HEAD_DIM; ++d) {
      float hv = ht[r * INNER + hd * HEAD_DIM + d];
      s1 += hv * attn[hd * 96 + d];
      s2 += hv * attn[hd * 96 + 32 + d];
    }
    ps[((size_t)row0 + r) * HEADS + hd] = s1;
    pd[((size_t)row0 + r) * HEADS + hd] = s2;
  }
}

// ---------------------------------------- scores + leaky relu + segment max
__global__ void k_edge_scores(const float* __restrict__ ea,
                              const int* __restrict__ eidx,
                              const float* __restrict__ Wedge,
                              const float* __restrict__ attn,
                              const float* __restrict__ ps,
                              const float* __restrict__ pd,
                              float* __restrict__ sc,
                              unsigned* __restrict__ mseg) {
  __shared__ float qe[HEADS * WEDGE];  // q_e[hd][w] = sum_d a_e[hd][d]*Wedge[hd*32+d][w]
  const int tid = threadIdx.x;
  if (tid < HEADS * WEDGE) {
    int hd = tid >> 4, w = tid & 15;
    float s = 0.0f;
    for (int d = 0; d < HEAD_DIM; ++d)
      s += attn[hd * 96 + 64 + d] * Wedge[(hd * HEAD_DIM + d) * WEDGE + w];
    qe[tid] = s;
  }
  __syncthreads();
  const long gt = (long)blockIdx.x * blockDim.x + tid;
  if (gt >= (long)BB * EE) return;
  const int b = (int)(gt / EE), e = (int)(gt % EE);
  const int srcn = eidx[e * 2 + 0];
  const int dstn = eidx[e * 2 + 1];
  float eav[WEDGE];
  for (int w = 0; w < WEDGE; ++w)
    eav[w] = san(ea[((size_t)b * EE + e) * WEDGE + w]);
  for (int hd = 0; hd < HEADS; ++hd) {
    float s = ps[((size_t)b * NN + srcn) * HEADS + hd] +
              pd[((size_t)b * NN + dstn) * HEADS + hd];
    for (int w = 0; w < WEDGE; ++w) s += eav[w] * qe[hd * WEDGE + w];
    s = (s >= 0.0f) ? s : NEG_SLOPE * s;
    sc[((size_t)b * EE + e) * HEADS + hd] = s;
    atomicMax(&mseg[((size_t)dstn * BB + b) * HEADS + hd], fenc(s));
  }
}

// ------------------------------------------- exp(s - m[dst]) + segment sum
__global__ void k_edge_softmax(const int* __restrict__ eidx,
                               const unsigned* __restrict__ mseg,
                               float* __restrict__ sc,
                               float* __restrict__ den) {
  const long gt = (long)blockIdx.x * blockDim.x + threadIdx.x;
  if (gt >= (long)BB * EE) return;
  const int b = (int)(gt / EE), e = (int)(gt % EE);
  const int dstn = eidx[e * 2 + 1];
  for (int hd = 0; hd < HEADS; ++hd) {
    size_t si = ((size_t)b * EE + e) * HEADS + hd;
    float m  = fdec(mseg[((size_t)dstn * BB + b) * HEADS + hd]);
    float ex = expf(sc[si] - m);
    sc[si] = ex;                                    // sc now holds numerator
    atomicAdd(&den[((size_t)dstn * BB + b) * HEADS + hd], ex);
  }
}

// ---------------- msg = gather(h,src) @ Wmsg^T, scaled by alpha, scatter-add
__global__ void k_msg_scatter(const float* __restrict__ h,
                              const int* __restrict__ eidx,
                              const float* __restrict__ Wmsg,
                              const float* __restrict__ sc,
                              const float* __restrict__ den,
                              float* __restrict__ agg) {
  __align__(16) __shared__ float at[16 * INNER];
  __shared__ float alpha_s[16 * HEADS];
  __shared__ int   dst_s[16];
  const int e0 = blockIdx.x * 16;
  const int b  = blockIdx.y;
  const int tid = threadIdx.x;
  // async gather of 16 h_src rows into LDS: 512 x b128 chunks, bypasses VGPRs
  {
    const unsigned at_base = (unsigned)(uintptr_t)(&at[0]);
    for (int i = 0; i < 2; ++i) {
      int li = tid + 256 * i;          // 0..511, 16 bytes each
      int r  = li >> 5;                // row (edge within tile)
      int c4 = li & 31;                // 16B chunk within 512B row
      int srcn = eidx[(e0 + r) * 2 + 0];
      async_copy_b128(at_base + (unsigned)li * 16u,
                      &h[((size_t)b * NN + srcn) * INNER + c4 * 4]);
    }
  }
  if (tid < 16) dst_s[tid] = eidx[(e0 + tid) * 2 + 1];
  if (tid < 16 * HEADS) {
    int r = tid >> 2, hd = tid & 3;
    int e = e0 + r;
    int dn = eidx[e * 2 + 1];
    alpha_s[r * HEADS + hd] = sc[((size_t)b * EE + e) * HEADS + hd] /
                              den[((size_t)dn * BB + b) * HEADS + hd];
  }
  async_wait0();
  __syncthreads();
  const int wave = tid >> 5, lane = tid & 31;
  v8f acc = wmma_tile(at, Wmsg, wave * 16, lane);
  const int hf  = lane >> 4;
  const int col = wave * 16 + (lane & 15);
  const int hd  = col >> 5;           // head = col / HEAD_DIM
  for (int v = 0; v < 8; ++v) {
    int rm = v + 8 * hf;
    float val = acc[v] * alpha_s[rm * HEADS + hd];
    atomicAdd(&agg[((size_t)b * NN + dst_s[rm]) * INNER + col], val);
  }
}

// -------------------------- out = agg @ Wout^T + x, then layernorm + sanitize
__global__ void k_out_ln(const float* __restrict__ agg,
                         const float* __restrict__ x,
                         const float* __restrict__ Wout,
                         const float* __restrict__ lng,
                         const float* __restrict__ lnb,
                         float* __restrict__ out) {
  __align__(16) __shared__ float at[16 * INNER];
  __shared__ float zt[16 * OUTD];
  __shared__ float reds[256], redq[256];
  __shared__ float mu_s[16], rs_s[16];
  const int row0 = blockIdx.x * 16;
  const int tid  = threadIdx.x;
  {
    const unsigned at_base = (unsigned)(uintptr_t)(&at[0]);
    for (int i = 0; i < 2; ++i) {
      int li = tid + 256 * i;          // 0..511, 16 bytes each
      async_copy_b128(at_base + (unsigned)li * 16u,
                      &agg[(size_t)row0 * INNER + (size_t)li * 4]);
    }
  }
  async_wait0();
  __syncthreads();
  const int wave = tid >> 5, lane = tid & 31;
  v8f acc = wmma_tile(at, Wout, wave * 16, lane);
  const int hf  = lane >> 4;
  const int col = wave * 16 + (lane & 15);
  for (int v = 0; v < 8; ++v) {
    int rm = v + 8 * hf;
    zt[rm * OUTD + col] = acc[v] + san(x[((size_t)row0 + rm) * FIN + col]);
  }
  __syncthreads();
  {   // 16 threads per row: partial mean / mean-of-squares over 8 cols each
    int r = tid >> 4, g = tid & 15;
    float s = 0.0f, q = 0.0f;
    for (int j = 0; j < 8; ++j) {
      float v = zt[r * OUTD + g * 8 + j];
      s += v; q += v * v;
    }
    reds[tid] = s; redq[tid] = q;
  }
  __syncthreads();
  for (int st = 8; st > 0; st >>= 1) {
    if ((tid & 15) < st) { reds[tid] += reds[tid + st]; redq[tid] += redq[tid + st]; }
    __syncthreads();
  }
  if ((tid & 15) == 0) {
    int r = tid >> 4;
    float mu  = reds[tid] * (1.0f / OUTD);
    float var = redq[tid] * (1.0f / OUTD) - mu * mu;
    mu_s[r] = mu;
    rs_s[r] = rsqrtf(var + LN_EPS);
  }
  __syncthreads();
  for (int i = 0; i < 8; ++i) {
    int li = tid + 256 * i;
    int r = li >> 7, c = li & 127;
    float y = (zt[r * OUTD + c] - mu_s[r]) * rs_s[r] * lng[c] + lnb[c];
    out[(size_t)row0 * OUTD + li] = san(y);
  }
}

extern "C" void kernel_launch(void* const* d_in, const int* in_sizes, int n_in,
                              void* d_out, int out_size, void* d_ws, size_t ws_size,
                              hipStream_t stream) {
  const float* x     = (const float*)d_in[0];
  const float* ea    = (const float*)d_in[1];
  const int*   eidx  = (const int*)d_in[2];
  const float* Wn    = (const float*)d_in[3];
  const float* Wedge = (const float*)d_in[4];
  const float* Wmsg  = (const float*)d_in[5];
  const float* attn  = (const float*)d_in[6];
  const float* Wout  = (const float*)d_in[7];
  const float* lng   = (const float*)d_in[8];
  const float* lnb   = (const float*)d_in[9];
  float* out = (float*)d_out;

  char* wsp = (char*)d_ws;
  auto carve = [&](size_t bytes) {
    char* p = wsp;
    wsp += (bytes + 255) & ~(size_t)255;
    return (void*)p;
  };
  float*    h    = (float*)   carve(sizeof(float) * BB * NN * INNER);
  float*    ps   = (float*)   carve(sizeof(float) * BB * NN * HEADS);
  float*    pd   = (float*)   carve(sizeof(float) * BB * NN * HEADS);
  float*    sc   = (float*)   carve(sizeof(float) * BB * EE * HEADS);
  unsigned* mseg = (unsigned*)carve(sizeof(unsigned) * NN * BB * HEADS);
  float*    den  = (float*)   carve(sizeof(float) * NN * BB * HEADS);
  float*    agg  = (float*)   carve(sizeof(float) * BB * NN * INNER);

  k_init<<<dim3((BB * NN * INNER + 255) / 256), dim3(256), 0, stream>>>(agg, den, mseg);
  k_node_transform<<<dim3(BB * NN / 16), dim3(256), 0, stream>>>(x, Wn, attn, h, ps, pd);
  k_edge_scores<<<dim3((BB * EE + 255) / 256), dim3(256), 0, stream>>>(ea, eidx, Wedge, attn,
                                                                       ps, pd, sc, mseg);
  k_edge_softmax<<<dim3((BB * EE + 255) / 256), dim3(256), 0, stream>>>(eidx, mseg, sc, den);
  k_msg_scatter<<<dim3(EE / 16, BB), dim3(256), 0, stream>>>(h, eidx, Wmsg, sc, den, agg);
  k_out_ln<<<dim3(BB * NN / 16), dim3(256), 0, stream>>>(agg, x, Wout, lng, lnb, out);
}